// TAN_56092272886361
// MI455X (gfx1250) — compile-verified
//
#include <hip/hip_runtime.h>
#include <math.h>

typedef __attribute__((ext_vector_type(2))) float v2f;
typedef __attribute__((ext_vector_type(8))) float v8f;

#define B_    16
#define C_    512
#define N_    64
#define K_    4096
#define TOPK_ 10

// ---------------------------------------------------------------------------
// Kernel 1: pred[b,k] = sum_c fused_h[b,c,k]*W[c] + bias; score = sigmoid*mask
// One wave per 16-wide k tile.  WMMA f32 16x16x4:
//   A(16x4)  = W[c..c+3] broadcast down all 16 rows
//   B(4x16)  = fused_h[b, c..c+3, k0..k0+15]
//   D[m,n]   = pred chunk, identical across rows m, column n = k0+n
// A layout (32-bit, 16x4): v0 = K0 (lanes0-15) / K2 (lanes16-31), v1 = K1/K3
// B layout (32-bit, 4x16): v0 = row K0 / row K2, v1 = row K1 / row K3
//  -> per lane: half = lane>=16 selects channel pair (c,c+1) vs (c+2,c+3)
// Two independent accumulator chains (channels [0,256) and [256,512)) for ILP.
// ---------------------------------------------------------------------------
__global__ __launch_bounds__(256) void tan_score_wmma(
    const float* __restrict__ fused_h,   // (B, C, K)
    const float* __restrict__ map_mask,  // (B, 1, N, N) -> (B, K)
    const float* __restrict__ W,         // (C)
    const float* __restrict__ bias,      // (1)
    float* __restrict__ score)           // (B, K) in workspace
{
    const int gwave = (blockIdx.x * 256 + threadIdx.x) >> 5;
    const int lane  = threadIdx.x & 31;
    const int batch = gwave >> 8;          // 256 tiles per batch
    const int tile  = gwave & 255;
    const int k0    = tile << 4;
    const int half  = lane >> 4;           // 0: channels c,c+1 ; 1: c+2,c+3
    const int col   = k0 + (lane & 15);

    const float* bp = fused_h + (size_t)batch * C_ * K_
                              + (size_t)(half * 2) * K_ + col;
    const float* wp = W + half * 2;

    v8f acc0 = {};
    v8f acc1 = {};
    #pragma unroll 4
    for (int c = 0; c < C_ / 2; c += 4) {
        const size_t o = (size_t)c * K_;
        // streamed exactly once -> non-temporal
        float d00 = __builtin_nontemporal_load(bp + o);
        float d01 = __builtin_nontemporal_load(bp + o + K_);
        float d10 = __builtin_nontemporal_load(bp + o + (size_t)(C_ / 2) * K_);
        float d11 = __builtin_nontemporal_load(bp + o + (size_t)(C_ / 2 + 1) * K_);
        v2f a0 = { wp[c],          wp[c + 1] };
        v2f a1 = { wp[c + C_ / 2], wp[c + C_ / 2 + 1] };
        v2f b0 = { d00, d01 };
        v2f b1 = { d10, d11 };
        acc0 = __builtin_amdgcn_wmma_f32_16x16x4_f32(
                   false, a0, false, b0, (short)0, acc0, false, false);
        acc1 = __builtin_amdgcn_wmma_f32_16x16x4_f32(
                   false, a1, false, b1, (short)0, acc1, false, false);
    }

    // D row 0 lives in acc[0]: lanes 0..15 hold columns n = 0..15 of the tile.
    if (lane < 16) {
        const int k  = k0 + lane;
        float pred   = acc0[0] + acc1[0] + bias[0];
        float s      = 1.0f / (1.0f + expf(-pred));
        float m      = map_mask[(size_t)batch * K_ + k];
        score[(size_t)batch * K_ + k] = s * m;
    }
}

// ---------------------------------------------------------------------------
// Kernel 2: per-batch stable descending sort (bitonic, LDS) + greedy NMS.
// Comparator encodes jnp.argsort(-score) stability: score desc, index asc.
// A box is suppressed iff it overlaps an earlier KEPT box (greedy NMS
// equivalence), so thread 0 scans the sorted list against <=10 kept boxes.
// ---------------------------------------------------------------------------
__global__ __launch_bounds__(256) void tan_nms(
    const float* __restrict__ score,     // (B, K)
    const float* __restrict__ se_times,  // (B, K, 2)
    float* __restrict__ out)             // (B, TOPK, 2)
{
    __shared__ float s_sc[K_];
    __shared__ int   s_ix[K_];
    const int batch = blockIdx.x;
    const float* sc = score + (size_t)batch * K_;

    for (int i = threadIdx.x; i < K_; i += 256) {
        s_sc[i] = sc[i];
        s_ix[i] = i;
    }

    // Bitonic sort: "before(p,i)" == p should precede i in final order.
    for (int kk = 2; kk <= K_; kk <<= 1) {
        for (int j = kk >> 1; j > 0; j >>= 1) {
            __syncthreads();
            for (int i = threadIdx.x; i < K_; i += 256) {
                const int p = i ^ j;
                if (p > i) {
                    float si = s_sc[i], sp = s_sc[p];
                    int   ii = s_ix[i], ip = s_ix[p];
                    bool pFirst = (sp > si) || (sp == si && ip < ii);
                    if (pFirst == ((i & kk) == 0)) {
                        s_sc[i] = sp; s_sc[p] = si;
                        s_ix[i] = ip; s_ix[p] = ii;
                    }
                }
            }
        }
    }
    __syncthreads();

    if (threadIdx.x == 0) {
        const float* se = se_times + (size_t)batch * K_ * 2;
        float ks[TOPK_], ke[TOPK_];
        int kc = 0;
        for (int p = 0; p < K_ && kc < TOPK_; ++p) {
            const int idx = s_ix[p];
            const float s1 = se[idx * 2], e1 = se[idx * 2 + 1];
            bool sup = false;
            for (int q = 0; q < kc; ++q) {
                float inter = fminf(e1, ke[q]) - fmaxf(s1, ks[q]);
                inter = fmaxf(inter, 0.0f);
                float uni = (e1 - s1) + (ke[q] - ks[q]) - inter;
                if (inter / (uni + 1e-9f) > 0.5f) { sup = true; break; }
            }
            if (!sup) { ks[kc] = s1; ke[kc] = e1; ++kc; }
        }
        // Rare fill path: fewer than TOPK kept -> kept set above is complete;
        // append suppressed boxes in sorted order (matches K+idxs ordering).
        if (kc < TOPK_) {
            int total = kc;
            for (int p = 0; p < K_ && total < TOPK_; ++p) {
                const int idx = s_ix[p];
                const float s1 = se[idx * 2], e1 = se[idx * 2 + 1];
                bool sup = false;
                for (int q = 0; q < kc; ++q) {
                    float inter = fminf(e1, ke[q]) - fmaxf(s1, ks[q]);
                    inter = fmaxf(inter, 0.0f);
                    float uni = (e1 - s1) + (ke[q] - ks[q]) - inter;
                    if (inter / (uni + 1e-9f) > 0.5f) { sup = true; break; }
                }
                if (sup) { ks[total] = s1; ke[total] = e1; ++total; }
            }
        }
        float* ob = out + batch * TOPK_ * 2;
        for (int t = 0; t < TOPK_; ++t) {
            ob[t * 2]     = ks[t];
            ob[t * 2 + 1] = ke[t];
        }
    }
}

extern "C" void kernel_launch(void* const* d_in, const int* in_sizes, int n_in,
                              void* d_out, int out_size, void* d_ws, size_t ws_size,
                              hipStream_t stream) {
    const float* fused_h  = (const float*)d_in[0];  // (16,512,64,64)
    const float* map_mask = (const float*)d_in[1];  // (16,1,64,64)
    const float* se_times = (const float*)d_in[2];  // (16,64,64,2)
    const float* W        = (const float*)d_in[3];  // (512)
    const float* bvec     = (const float*)d_in[4];  // (1)
    float* score = (float*)d_ws;                    // (16,4096) = 256 KB

    // B*K/16 = 4096 wave-tiles, 8 waves per 256-thread block -> 512 blocks.
    tan_score_wmma<<<512, 256, 0, stream>>>(fused_h, map_mask, W, bvec, score);
    tan_nms<<<B_, 256, 0, stream>>>(score, se_times, (float*)d_out);
}